// UncertainTemporalMemory_76836964926233
// MI455X (gfx1250) — compile-verified
//
#include <hip/hip_runtime.h>

#define DIM 128          // floats per row
#define LANES 32         // wave32: one wave owns one row, 4 floats per lane
#define MOM 0.9f
#define OneM 0.1f        // 1 - momentum
#define PF_ROWS 48       // prefetch distance (rows ahead) for streaming copy

typedef float v4f __attribute__((ext_vector_type(4)));

// ---------------------------------------------------------------------------
// HOT PASS (placed first so the disasm snippet shows its codegen):
// streaming copy of memory/variance -> new_memory/new_variance, fused with
// the EMA update on rows that have a winning batch element.
// One wave32 per node row: lane L moves floats [4L, 4L+4) as one B128.
__global__ void utm_copy_update(const float* __restrict__ mem,
                                const float* __restrict__ var,
                                const float* __restrict__ vals,
                                const int*   __restrict__ owner,
                                float* __restrict__ out_nm,
                                float* __restrict__ out_nv,
                                int N) {
    unsigned t = blockIdx.x * blockDim.x + threadIdx.x;
    int row  = (int)(t >> 5);
    int lane = (int)(t & 31);
    if (row >= N) return;

    size_t off = (size_t)row * DIM + (size_t)lane * 4;

    // Prefetch the streaming inputs well ahead (global_prefetch_b8).
    if (row + PF_ROWS < N) {
        __builtin_prefetch(mem + off + (size_t)PF_ROWS * DIM, 0, 0);
        __builtin_prefetch(var + off + (size_t)PF_ROWS * DIM, 0, 0);
    }

    // Streamed once -> non-temporal loads (don't pollute L2).
    v4f mv = __builtin_nontemporal_load((const v4f*)(mem + off));
    v4f vv = __builtin_nontemporal_load((const v4f*)(var + off));

    int b = owner[row];   // wave-uniform (same address across the wave)
    if (b >= 0) {
        size_t voff = (size_t)b * DIM + (size_t)lane * 4;
        v4f val = *(const v4f*)(vals + voff);
        v4f d   = val - mv;
        v4f nv  = MOM * vv + OneM * d * d;
        v4f nm  = MOM * mv + OneM * val;
        mv = nm; vv = nv;
    }

    // Written once, never re-read -> non-temporal stores.
    __builtin_nontemporal_store(mv, (v4f*)(out_nm + off));
    __builtin_nontemporal_store(vv, (v4f*)(out_nv + off));
}

// ---------------------------------------------------------------------------
// Pass 1a: owner[i] = -1
__global__ void utm_init_owner(int* __restrict__ owner, int n) {
    int i = blockIdx.x * blockDim.x + threadIdx.x;
    if (i < n) owner[i] = -1;
}

// Pass 1b: owner[node] = max batch index writing that node (deterministic
// duplicate resolution: "last occurrence wins", matching fancy-index set).
__global__ void utm_build_owner(const int* __restrict__ idx,
                                int* __restrict__ owner, int B) {
    int b = blockIdx.x * blockDim.x + threadIdx.x;
    if (b < B) atomicMax(&owner[idx[b]], b);
}

// ---------------------------------------------------------------------------
// Pass 3: per-batch-row gather of the OLD memory/variance (pure gather; reads
// the immutable inputs, so it's independent of pass 2).
__global__ void utm_gather(const int* __restrict__ idx,
                           const float* __restrict__ mem,
                           const float* __restrict__ var,
                           float* __restrict__ out_mr,
                           float* __restrict__ out_vr,
                           int B) {
    unsigned t = blockIdx.x * blockDim.x + threadIdx.x;
    int b    = (int)(t >> 5);
    int lane = (int)(t & 31);
    if (b >= B) return;

    int node = idx[b];
    size_t soff = (size_t)node * DIM + (size_t)lane * 4;
    size_t doff = (size_t)b    * DIM + (size_t)lane * 4;

    v4f mv = *(const v4f*)(mem + soff);
    v4f vv = *(const v4f*)(var + soff);

    __builtin_nontemporal_store(mv, (v4f*)(out_mr + doff));
    __builtin_nontemporal_store(vv, (v4f*)(out_vr + doff));
}

// ---------------------------------------------------------------------------
// Fallback (no workspace for owner map): plain copy, then scatter update.
// Duplicate idx resolution is whichever scatter lands last ("unspecified
// winner" semantics, matching the reference, but not bit-deterministic).
__global__ void utm_plain_copy(const float* __restrict__ mem,
                               const float* __restrict__ var,
                               float* __restrict__ out_nm,
                               float* __restrict__ out_nv,
                               int N) {
    unsigned t = blockIdx.x * blockDim.x + threadIdx.x;
    int row  = (int)(t >> 5);
    int lane = (int)(t & 31);
    if (row >= N) return;
    size_t off = (size_t)row * DIM + (size_t)lane * 4;
    v4f mv = __builtin_nontemporal_load((const v4f*)(mem + off));
    v4f vv = __builtin_nontemporal_load((const v4f*)(var + off));
    __builtin_nontemporal_store(mv, (v4f*)(out_nm + off));
    __builtin_nontemporal_store(vv, (v4f*)(out_nv + off));
}

__global__ void utm_scatter_update(const int* __restrict__ idx,
                                   const float* __restrict__ mem,
                                   const float* __restrict__ var,
                                   const float* __restrict__ vals,
                                   float* __restrict__ out_mr,
                                   float* __restrict__ out_vr,
                                   float* __restrict__ out_nm,
                                   float* __restrict__ out_nv,
                                   int B) {
    unsigned t = blockIdx.x * blockDim.x + threadIdx.x;
    int b    = (int)(t >> 5);
    int lane = (int)(t & 31);
    if (b >= B) return;

    int node = idx[b];
    size_t soff = (size_t)node * DIM + (size_t)lane * 4;
    size_t doff = (size_t)b    * DIM + (size_t)lane * 4;

    v4f mv  = *(const v4f*)(mem  + soff);
    v4f vv  = *(const v4f*)(var  + soff);
    v4f val = *(const v4f*)(vals + doff);

    __builtin_nontemporal_store(mv, (v4f*)(out_mr + doff));
    __builtin_nontemporal_store(vv, (v4f*)(out_vr + doff));

    v4f d  = val - mv;
    v4f nv = MOM * vv + OneM * d * d;
    v4f nm = MOM * mv + OneM * val;

    *(v4f*)(out_nm + soff) = nm;
    *(v4f*)(out_nv + soff) = nv;
}

// ---------------------------------------------------------------------------
extern "C" void kernel_launch(void* const* d_in, const int* in_sizes, int n_in,
                              void* d_out, int out_size, void* d_ws, size_t ws_size,
                              hipStream_t stream) {
    const int*   idx      = (const int*)  d_in[0];
    const float* values   = (const float*)d_in[1];
    const float* memory   = (const float*)d_in[2];
    const float* variance = (const float*)d_in[3];

    const int B = in_sizes[0];
    const int N = in_sizes[2] / DIM;

    float* out     = (float*)d_out;
    float* out_mr  = out;                              // mem_read    [B, D]
    float* out_vr  = out_mr + (size_t)B * DIM;         // var_read    [B, D]
    float* out_nm  = out_vr + (size_t)B * DIM;         // new_memory  [N, D]
    float* out_nv  = out_nm + (size_t)N * DIM;         // new_variance[N, D]

    const int TPB = 256;   // 8 wave32s per block
    const unsigned rowThreadsN = (unsigned)N * LANES;
    const unsigned rowThreadsB = (unsigned)B * LANES;

    if (ws_size >= (size_t)N * sizeof(int)) {
        int* owner = (int*)d_ws;
        utm_init_owner <<<(N + TPB - 1) / TPB, TPB, 0, stream>>>(owner, N);
        utm_build_owner<<<(B + TPB - 1) / TPB, TPB, 0, stream>>>(idx, owner, B);
        utm_copy_update<<<(rowThreadsN + TPB - 1) / TPB, TPB, 0, stream>>>(
            memory, variance, values, owner, out_nm, out_nv, N);
        utm_gather<<<(rowThreadsB + TPB - 1) / TPB, TPB, 0, stream>>>(
            idx, memory, variance, out_mr, out_vr, B);
    } else {
        utm_plain_copy<<<(rowThreadsN + TPB - 1) / TPB, TPB, 0, stream>>>(
            memory, variance, out_nm, out_nv, N);
        utm_scatter_update<<<(rowThreadsB + TPB - 1) / TPB, TPB, 0, stream>>>(
            idx, memory, variance, values, out_mr, out_vr, out_nm, out_nv, B);
    }
}